// RegLossSparse_18227841204812
// MI455X (gfx1250) — compile-verified
//
#include <hip/hip_runtime.h>

typedef __attribute__((ext_vector_type(2))) float v2f;
typedef __attribute__((ext_vector_type(8))) float v8f;

#define NROWS    2000   // B * MAXOBJ
#define BATCHES  4
#define MPB      500
#define DIM      8
#define NTHREADS 256
#define NWAVES   8
#define NCHUNKS  250    // NROWS / 8 rows per WMMA

// Wave-parallel lower_bound over a sorted int array: 32 probes split the
// bracket into 33 segments per round (~4 dependent-load rounds for 4M elems
// instead of ~22 for scalar binary search). All 32 lanes must be active.
__device__ __forceinline__ int wave_lower_bound(const int* __restrict__ arr,
                                                int n, int tgt, int lane) {
  int lo = 0, hi = n;                       // answer in [lo, hi]
  while (hi - lo > 32) {
    long long span = (long long)(hi - lo);
    int pos = lo + (int)((span * (long long)(lane + 1)) / 33);
    int v = arr[pos];
    unsigned long long bal = __ballot(v < tgt);
    int cnt = __popcll(bal);                // contiguous low bits (sorted)
    int nlo = (cnt == 0)  ? lo : (lo + (int)((span * (long long)cnt) / 33) + 1);
    int nhi = (cnt == 32) ? hi : (lo + (int)((span * (long long)(cnt + 1)) / 33));
    lo = nlo; hi = nhi;
  }
  int pos = lo + lane;
  int v = (pos < hi) ? arr[pos] : 0x7fffffff;
  unsigned long long bal = __ballot(v < tgt);
  return lo + __popcll(bal);
}

__global__ __launch_bounds__(NTHREADS)
void RegLossSparse_kernel(const float* __restrict__ output,
                          const unsigned char* __restrict__ mask,   // jnp.bool_ = 1 byte
                          const int* __restrict__ ind,
                          const float* __restrict__ target,
                          const int* __restrict__ batch_index,
                          int n_points,
                          float* __restrict__ out) {
  __shared__ int   s_starts[BATCHES + 1];
  __shared__ int   s_gidx[NROWS];   // -1 => empty batch (pred forced to 0)
  __shared__ int   s_m[NROWS];      // mask bit per row
  __shared__ int   s_num;
  __shared__ float s_part[NWAVES * 16];

  const int tid  = threadIdx.x;
  const int wave = tid >> 5;
  const int lane = tid & 31;

  if (tid == 0) { s_starts[0] = 0; s_starts[BATCHES] = n_points; s_num = 0; }

  // ---- Phase 1a: batch start offsets via wave-parallel lower_bound --------
  if (wave >= 1 && wave <= BATCHES - 1) {
    int r = wave_lower_bound(batch_index, n_points, /*tgt=*/wave, lane);
    if (lane == 0) s_starts[wave] = r;
  }
  __syncthreads();

  // ---- Phase 1b: per-row gather index + mask, count num -------------------
  int mynum = 0;
  for (int r = tid; r < NROWS; r += NTHREADS) {
    int b   = r / MPB;
    int st  = s_starts[b];
    int cnt = s_starts[b + 1] - st;
    int iv  = ind[r];
    int him = max(cnt, 1) - 1;                 // clip(ind, 0, max(cnt,1)-1)
    iv = min(max(iv, 0), him);
    s_gidx[r] = (cnt == 0) ? -1 : (st + iv);
    int mb = mask[r] ? 1 : 0;
    s_m[r] = mb;
    mynum += mb;
  }
  atomicAdd(&s_num, mynum);
  __syncthreads();

  // ---- Phase 2: WMMA-accumulated masked L1 reduction ----------------------
  // B matrix (4x16 f32): VGPR0 holds rows K={0,2}, VGPR1 rows K={1,3};
  // N striped across lane halves. Column n = dim + 8*sub; chunk rows:
  // sub=0 -> rows r0..r0+3 (cols 0..7), sub=1 -> rows r0+4..r0+7 (cols 8..15).
  const int n   = lane & 15;
  const int d   = n & 7;
  const int sub = n >> 3;
  const int kb  = (lane >= 16) ? 2 : 0;
  const int ro0 = kb + 4 * sub;       // row offset for B vgpr0 element
  const int ro1 = kb + 1 + 4 * sub;   // row offset for B vgpr1 element

  v2f a; a.x = 1.0f; a.y = 1.0f;      // all-ones A (16x4): D = column sums of B
  v8f acc = {};

  for (int c = wave; c < NCHUNKS; c += NWAVES) {
    int r0 = c * 8;
    v2f bv;
    {
      int row = r0 + ro0;
      int g   = s_gidx[row];
      int gs  = (g < 0) ? 0 : g;
      float p = output[(size_t)gs * DIM + d];
      p = (g < 0) ? 0.0f : p;                          // empty batch => pred 0
      float t  = target[row * DIM + d];
      bool ok  = (s_m[row] != 0) && !(p != p) && !(t != t);
      bv.x = ok ? fabsf(p - t) : 0.0f;
    }
    {
      int row = r0 + ro1;
      int g   = s_gidx[row];
      int gs  = (g < 0) ? 0 : g;
      float p = output[(size_t)gs * DIM + d];
      p = (g < 0) ? 0.0f : p;
      float t  = target[row * DIM + d];
      bool ok  = (s_m[row] != 0) && !(p != p) && !(t != t);
      bv.y = ok ? fabsf(p - t) : 0.0f;
    }
    // v_wmma_f32_16x16x4_f32: acc += ones(16x4) x bv(4x16)
    acc = __builtin_amdgcn_wmma_f32_16x16x4_f32(
        /*neg_a=*/false, a, /*neg_b=*/false, bv,
        /*c_mod=*/(short)0, acc, /*reuse_a=*/false, /*reuse_b=*/false);
  }

  // D row M=0 lives in acc[0] of lanes 0..15 (N = lane).
  if (lane < 16) s_part[wave * 16 + lane] = acc[0];
  __syncthreads();

  // ---- Phase 3: fold waves + column halves, normalize ---------------------
  if (tid < DIM) {
    float s = 0.0f;
    for (int w = 0; w < NWAVES; ++w)
      s += s_part[w * 16 + tid] + s_part[w * 16 + tid + 8];
    float denom = fmaxf((float)s_num, 1.0f);
    float v = s / denom;
    out[tid] = (v != v) ? 0.0f : v;
  }
}

extern "C" void kernel_launch(void* const* d_in, const int* in_sizes, int n_in,
                              void* d_out, int out_size, void* d_ws, size_t ws_size,
                              hipStream_t stream) {
  const float*         output      = (const float*)d_in[0];
  const unsigned char* mask        = (const unsigned char*)d_in[1];
  const int*           ind         = (const int*)d_in[2];
  const float*         target      = (const float*)d_in[3];
  const int*           batch_index = (const int*)d_in[4];
  float*               out         = (float*)d_out;
  const int n_points = in_sizes[4];   // N

  RegLossSparse_kernel<<<1, NTHREADS, 0, stream>>>(
      output, mask, ind, target, batch_index, n_points, out);
}